// ElementLoss_77043123355910
// MI455X (gfx1250) — compile-verified
//
#include <hip/hip_runtime.h>
#include <math.h>

typedef __attribute__((ext_vector_type(2))) float v2f;
typedef __attribute__((ext_vector_type(8))) float v8f;

#define DIMS 128
#define NPART 1024
// ws layout (floats): [0..NPART-1] sim_loss partials, [NPART] sumsq(C), [NPART+1] sumsq(A)

__device__ __forceinline__ float wave_sum(float v) {
#pragma unroll
  for (int o = 16; o > 0; o >>= 1) v += __shfl_xor(v, o, 32);
  return v;
}

__global__ void init_ws(float* ws) {
  int t = blockIdx.x * blockDim.x + threadIdx.x;
  if (t < NPART + 2) ws[t] = 0.f;
}

__global__ void sumsq_kernel(const float* __restrict__ v, int n, float* __restrict__ out) {
  int stride = gridDim.x * blockDim.x;
  float s = 0.f;
  for (int idx = blockIdx.x * blockDim.x + threadIdx.x; idx < n; idx += stride) {
    float x = v[idx];
    s += x * x;
  }
  s = wave_sum(s);
  if ((threadIdx.x & 31) == 0) atomicAdd(out, s);
}

// One wave processes 16 edges per group iteration.
// Half-wave lane m (m = lane & 15) owns edge (g*16 + m).
// Per K-chunk q: low lanes hold masked residual cols (4q, 4q+1), high lanes (4q+2, 4q+3).
// This register pattern is simultaneously a valid 16x4 A-matrix and (as B) its own
// transpose 4x16, so wmma_f32_16x16x4_f32(a, a, c) accumulates M*M^T: the diagonal
// of the 16x16 f32 accumulator is the exact masked squared L2 norm per edge.
__global__ void __launch_bounds__(256) edge_kernel(
    const float* __restrict__ X, const float* __restrict__ C, const float* __restrict__ A,
    const int* __restrict__ ei, const int* __restrict__ ek, const int* __restrict__ ej,
    const int* __restrict__ et, const float* __restrict__ ew, const float* __restrict__ es,
    const unsigned char* __restrict__ mask, float* __restrict__ ws, int E) {
  const int lane = threadIdx.x & 31;
  const int m = lane & 15;
  const int wavesPerBlock = blockDim.x >> 5;
  const int waveGlobal = blockIdx.x * wavesPerBlock + (threadIdx.x >> 5);
  const int nWaves = gridDim.x * wavesPerBlock;
  const int nGroups = (E + 15) >> 4;
  const int colBase = (lane < 16) ? 0 : 2;

  float acc = 0.f;
  for (int g = waveGlobal; g < nGroups; g += nWaves) {   // wave-uniform loop: EXEC all-1 at WMMA
    const long long e = (long long)g * 16 + m;
    const bool valid = e < (long long)E;
    const int ii = valid ? ei[e] : 0;
    const int kk = valid ? ek[e] : 0;
    const int jj = valid ? ej[e] : -1;
    const int tt = valid ? et[e] : 0;
    const float wgt = valid ? (ew[e] * es[e]) : 0.f;
    const float Ci = C[ii];
    const float cdiff = C[kk] - Ci;   // (X[k]+C[k]) - (X[i]+C[i]) scalar part
    const float delta = A[tt] - Ci;   // subtracted at column j before squaring
    const float* __restrict__ xi = X + (long long)ii * DIMS;
    const float* __restrict__ xk = X + (long long)kk * DIMS;
    const unsigned char* __restrict__ mrow = mask + e * DIMS;

    // keep the HBM mask stream ahead (global_prefetch_b8)
    {
      long long e2 = (long long)(g + nWaves) * 16 + m;
      if (e2 < (long long)E) __builtin_prefetch(mask + e2 * DIMS, 0, 0);
    }

    v8f c8 = {0.f, 0.f, 0.f, 0.f, 0.f, 0.f, 0.f, 0.f};
#pragma unroll
    for (int q = 0; q < DIMS / 4; ++q) {
      const int c0 = q * 4 + colBase;                    // even -> 8B/2B aligned
      const float2 xkv = *(const float2*)(xk + c0);
      const float2 xiv = *(const float2*)(xi + c0);
      const unsigned int mm =
          valid ? (unsigned int)(*(const unsigned short*)(mrow + c0)) : 0u;
      float a0 = xkv.x - xiv.x + cdiff;
      float a1 = xkv.y - xiv.y + cdiff;
      if (c0 == jj)     a0 -= delta;
      if (c0 + 1 == jj) a1 -= delta;
      a0 = (mm & 0x00ffu) ? a0 : 0.f;                    // masked_select
      a1 = (mm & 0xff00u) ? a1 : 0.f;
      v2f ab = {a0, a1};
      // D = A * A^T + C  (exact fp32, RNE); diagonal accumulates sum of squares
      c8 = __builtin_amdgcn_wmma_f32_16x16x4_f32(false, ab, false, ab,
                                                 (short)0, c8, false, false);
    }

    // Extract diagonal D[m,m]: m<8 -> (VGPR m, lane m); m>=8 -> (VGPR m-8, lane m+16)
    const int src = (m < 8) ? m : (m + 16);
    float sq = 0.f;
#pragma unroll
    for (int r = 0; r < 8; ++r) {
      float gv = __shfl(c8[r], src, 32);
      if ((m & 7) == r) sq = gv;
    }
    const float nrm = (sq > 0.f) ? sqrtf(sq) : 0.f;      // zero-safe norm
    const float contrib = (lane < 16) ? (wgt * nrm) : 0.f; // high half is a duplicate
    acc += wave_sum(contrib);
  }
  if (lane == 0) atomicAdd(&ws[waveGlobal & (NPART - 1)], acc);
}

__global__ void finalize_kernel(const float* __restrict__ ws, float* __restrict__ out) {
  __shared__ float red[8];
  float s = 0.f;
  for (int i = threadIdx.x; i < NPART; i += blockDim.x) s += ws[i];
  s = wave_sum(s);
  if ((threadIdx.x & 31) == 0) red[threadIdx.x >> 5] = s;
  __syncthreads();
  if (threadIdx.x == 0) {
    float tot = 0.f;
    for (int i = 0; i < (int)(blockDim.x >> 5); ++i) tot += red[i];
    out[0] = tot + sqrtf(ws[NPART]) + sqrtf(ws[NPART + 1]);
  }
}

extern "C" void kernel_launch(void* const* d_in, const int* in_sizes, int n_in,
                              void* d_out, int out_size, void* d_ws, size_t ws_size,
                              hipStream_t stream) {
  const float* X = (const float*)d_in[0];
  const float* C = (const float*)d_in[1];
  const float* A = (const float*)d_in[2];
  const int* ei = (const int*)d_in[3];
  const int* ek = (const int*)d_in[4];
  const int* ej = (const int*)d_in[5];
  const int* et = (const int*)d_in[6];
  const float* ew = (const float*)d_in[7];
  const float* es = (const float*)d_in[8];
  const unsigned char* mask = (const unsigned char*)d_in[9];
  const int T = in_sizes[1];
  const int NA = in_sizes[2];
  const int E = in_sizes[3];
  float* ws = (float*)d_ws;
  float* out = (float*)d_out;

  init_ws<<<(NPART + 2 + 255) / 256, 256, 0, stream>>>(ws);
  sumsq_kernel<<<64, 256, 0, stream>>>(C, T, ws + NPART);
  sumsq_kernel<<<1024, 256, 0, stream>>>(A, NA, ws + NPART + 1);
  edge_kernel<<<2048, 256, 0, stream>>>(X, C, A, ei, ek, ej, et, ew, es, mask, ws, E);
  finalize_kernel<<<1, 256, 0, stream>>>(ws, out);
}